// Clas_21912923144536
// MI455X (gfx1250) — compile-verified
//
#include <hip/hip_runtime.h>
#include <hip/hip_bf16.h>
#include <math.h>
#include <stdint.h>

// Row length cap (reference uses T = 32768 -> 128 KB, fits CDNA5's 320 KB WGP LDS).
#define TMAX   32768
#define NT     256   // 8 wave32 per workgroup
#define NWAVES 8

// One workgroup per video row:
//   1) async-DMA the valid prefix scores[row, 0:seqlen] into LDS (gfx1250 ASYNC path)
//   2) exact top-k sum via 4x8-bit radix select on the f32 bit patterns,
//      b128 LDS reads + per-wave histograms to kill atomic contention
//   3) emit per-row BCE term  label*log(p) + (1-label)*log1p(-p)
__global__ __launch_bounds__(NT) void topk_bce_row_kernel(
    const float* __restrict__ scores,
    const float* __restrict__ label,
    const int*   __restrict__ seqlen,
    float*       __restrict__ row_term,
    int T)
{
    __shared__ __align__(16) float s_data[TMAX];     // 128 KB row buffer
    __shared__ unsigned s_cnt[NWAVES][256];          // per-wave histogram counts (8 KB)
    __shared__ float    s_sum[NWAVES][256];          // per-wave histogram sums   (8 KB)
    __shared__ unsigned s_prefix;
    __shared__ unsigned s_krem;
    __shared__ float    s_total;

    const int row = blockIdx.x;
    const int tid = (int)threadIdx.x;
    const int wv  = tid >> 5;           // wave id 0..7
    const int n   = seqlen[row];        // 1 <= n <= T
    const int k   = (n >> 4) + 1;       // per-sample top-k, k <= n

    const float* src = scores + (size_t)row * (size_t)T;

    // ---- Stream the valid prefix into LDS with async-to-LDS DMA (ASYNCcnt) ----
    // Low 32 bits of a flat shared pointer are the LDS byte address.
    const unsigned lds_base = (unsigned)(uintptr_t)(void*)&s_data[0];
    const int nvec = (n + 3) >> 2;      // number of 16-byte vectors to fetch
    for (int v = tid; v < nvec; v += NT) {
        unsigned lds_off = lds_base + 16u * (unsigned)v;
        uint64_t ga      = (uint64_t)(uintptr_t)(src + 4 * v);
        asm volatile("global_load_async_to_lds_b128 %0, %1, off"
                     :: "v"(lds_off), "v"(ga)
                     : "memory");
    }
#if __has_builtin(__builtin_amdgcn_s_wait_asynccnt)
    __builtin_amdgcn_s_wait_asynccnt(0);
#else
    asm volatile("s_wait_asynccnt 0" ::: "memory");
#endif
    __syncthreads();   // all waves' DMA'd data now visible in LDS

    if (tid == 0) { s_prefix = 0u; s_krem = (unsigned)k; s_total = 0.0f; }

    const uint4* sv  = (const uint4*)s_data;   // 16B-aligned vector view
    const int    nv4 = n >> 2;
    const int    tail = n & 3;

    // ---- 4-pass radix select (MSB-first) over positive-float bit patterns ----
    for (int pass = 0; pass < 4; ++pass) {
        const int      shift = 24 - 8 * pass;
        const unsigned msk   = (pass == 0) ? 0u : (0xFFFFFFFFu << (32 - 8 * pass));

        // clear all per-wave histogram copies (256 bins each)
        {
            unsigned* pc = &s_cnt[0][0];
            float*    ps = &s_sum[0][0];
            #pragma unroll
            for (int i = 0; i < NWAVES; ++i) {
                pc[i * NT + tid] = 0u;
                ps[i * NT + tid] = 0.0f;
            }
        }
        __syncthreads();

        const unsigned pfx = s_prefix;
        unsigned* mycnt = &s_cnt[wv][0];
        float*    mysum = &s_sum[wv][0];

        // vectorized scan: ds_load_b128 per 4 elements
        for (int i = tid; i < nv4; i += NT) {
            uint4 q = sv[i];
            #pragma unroll
            for (int c = 0; c < 4; ++c) {
                unsigned u = (c == 0) ? q.x : (c == 1) ? q.y : (c == 2) ? q.z : q.w;
                if ((u & msk) == pfx) {
                    unsigned d = (u >> shift) & 255u;
                    atomicAdd(&mycnt[d], 1u);                 // ds_add_u32
                    atomicAdd(&mysum[d], __uint_as_float(u)); // ds_add_f32
                }
            }
        }
        // tail (< 4 elements)
        if (tid < tail) {
            unsigned u = __float_as_uint(s_data[4 * nv4 + tid]);
            if ((u & msk) == pfx) {
                unsigned d = (u >> shift) & 255u;
                atomicAdd(&mycnt[d], 1u);
                atomicAdd(&mysum[d], __uint_as_float(u));
            }
        }
        __syncthreads();

        // fold the 8 per-wave copies: thread tid owns bin tid
        {
            unsigned c = 0u;
            float    s = 0.0f;
            #pragma unroll
            for (int w = 0; w < NWAVES; ++w) {
                c += s_cnt[w][tid];
                s += s_sum[w][tid];
            }
            s_cnt[0][tid] = c;
            s_sum[0][tid] = s;
        }
        __syncthreads();

        if (tid == 0) {
            unsigned kr  = s_krem;
            float    tt  = s_total;
            unsigned cum = 0u;
            int d = 255;
            for (; d >= 0; --d) {
                unsigned c = s_cnt[0][d];
                if (cum + c >= kr) break;    // k-th largest lives in bin d
                cum += c;
                tt  += s_sum[0][d];          // bins above d are fully in top-k
            }
            if (d < 0) d = 0;                // unreachable (invariant: candidates >= kr)
            s_krem   = kr - cum;
            s_total  = tt;
            s_prefix = pfx | ((unsigned)d << shift);
        }
        __syncthreads();
    }

    if (tid == 0) {
        // Remaining k_rem candidates all equal the full 32-bit threshold value.
        float thr  = __uint_as_float(s_prefix);
        float tsum = s_total + (float)s_krem * thr;
        float p    = tsum / (float)k;        // mean of top-k == vl_scores[row]
        float lab  = label[row];
        row_term[row] = lab * logf(p) + (1.0f - lab) * log1pf(-p);
    }
}

// Reduce per-row BCE terms -> scalar loss = -mean(term)
__global__ __launch_bounds__(256) void bce_reduce_kernel(
    const float* __restrict__ row_term, float* __restrict__ out, int b)
{
    __shared__ float sh[256];
    float acc = 0.0f;
    for (int i = (int)threadIdx.x; i < b; i += 256) acc += row_term[i];
    sh[threadIdx.x] = acc;
    __syncthreads();
    for (int s = 128; s > 0; s >>= 1) {
        if ((int)threadIdx.x < s) sh[threadIdx.x] += sh[threadIdx.x + s];
        __syncthreads();
    }
    if (threadIdx.x == 0) out[0] = -sh[0] / (float)b;
}

extern "C" void kernel_launch(void* const* d_in, const int* in_sizes, int n_in,
                              void* d_out, int out_size, void* d_ws, size_t ws_size,
                              hipStream_t stream) {
    (void)n_in; (void)out_size; (void)ws_size;
    const float* scores = (const float*)d_in[0];   // [B, T] f32
    const float* label  = (const float*)d_in[1];   // [B]    f32
    const int*   seqlen = (const int*)d_in[2];     // [B]    i32

    const int b = in_sizes[1];
    const int T = in_sizes[0] / b;                 // == 32768 (<= TMAX)

    float* row_term = (float*)d_ws;                // B floats of scratch

    topk_bce_row_kernel<<<b, NT, 0, stream>>>(scores, label, seqlen, row_term, T);
    bce_reduce_kernel<<<1, 256, 0, stream>>>(row_term, (float*)d_out, b);
}